// NgramGPUKernel_40896678593078
// MI455X (gfx1250) — compile-verified
//
#include <hip/hip_runtime.h>
#include <hip/hip_bf16.h>

// Problem constants from the reference
#define BATCH   512
#define SEQ     8192
#define KOUT    8
#define NTHREADS 256
#define PPT      4          // positions evaluated per thread per iteration
#define IMAX     0x7fffffff

typedef unsigned int u32x4 __attribute__((ext_vector_type(4)));
typedef int          i32x4 __attribute__((ext_vector_type(4)));
typedef int          i32x8 __attribute__((ext_vector_type(8)));

__global__ __launch_bounds__(NTHREADS)
void ngram_match_extract_kernel(const int* __restrict__ num_tokens,
                                const int* __restrict__ tokens,
                                const unsigned char* __restrict__ mask,
                                int* __restrict__ out)
{
    __shared__ __align__(16) int s_tok[SEQ + 8];  // +8 pad so 8-wide windows never read OOB
    __shared__ int s_pat[5];
    __shared__ int s_min[3];                      // earliest match pos for n=3,4,5

    const int b   = blockIdx.x;
    const int tid = threadIdx.x;
    const long long row = (long long)b * SEQ;

    if (tid < 8) s_tok[SEQ + tid] = -1;           // sentinel: never equals a token (<32)
    if (tid < 3) s_min[tid] = IMAX;

    const int seq = num_tokens[b];

    // pattern = last 5 tokens ending at seq (indices clipped at 0, as in reference)
    if (tid < 5) {
        int idx = seq - 5 + tid;
        if (idx < 0) idx = 0;
        s_pat[tid] = tokens[row + idx];
    }

    // ---- Stage the row into LDS ------------------------------------------
#if defined(__has_builtin) && __has_builtin(__builtin_amdgcn_tensor_load_to_lds)
    // CDNA5 Tensor Data Mover: one DMA descriptor moves the whole 32KB row.
    if (tid < 32) {                              // single wave issues the TDM op
        const unsigned           lds_addr = (unsigned)(size_t)(void*)&s_tok[0];
        const unsigned long long ga       = (unsigned long long)(size_t)(tokens + row);
        u32x4 g0; i32x8 g1; i32x4 g2 = {0, 0, 0, 0}; i32x4 g3 = {0, 0, 0, 0};
        i32x8 g4 = {0, 0, 0, 0, 0, 0, 0, 0};
        // D# group 0: count=1 | lds_addr | global_addr[56:0] | type=2("image")
        g0.x = 1u;
        g0.y = lds_addr;
        g0.z = (unsigned)(ga & 0xffffffffu);
        g0.w = (unsigned)((ga >> 32) & 0x1ffffffu) | (2u << 30);
        // D# group 1: data_size=4B, tensor_dim0=SEQ, tensor_dim1=1,
        //             tile_dim0=SEQ, tile_dim1=1, dim0_stride=SEQ
        g1[0] = (int)(2u << 16);                          // data_size=2 (4 bytes), wg_mask=0
        g1[1] = (int)(((unsigned)SEQ & 0xffffu) << 16);   // tensor_dim0[15:0]
        g1[2] = (int)(((unsigned)SEQ >> 16) | (1u << 16));// tensor_dim0[31:16] | tensor_dim1[15:0]=1
        g1[3] = (int)(((unsigned)SEQ & 0xffffu) << 16);   // tensor_dim1[31:16]=0 | tile_dim0=SEQ
        g1[4] = 1;                                        // tile_dim1=1, tile_dim2=0
        g1[5] = SEQ;                                      // tensor_dim0_stride[31:0]
        g1[6] = 0;
        g1[7] = 0;
        // clang-23 / therock-10.0 signature: (g0, g1, g2, g3, g4, cpol)
        __builtin_amdgcn_tensor_load_to_lds(g0, g1, g2, g3, g4, 0);
        __builtin_amdgcn_s_wait_tensorcnt(0);
    }
#else
    for (int i = tid; i < SEQ / 4; i += NTHREADS)
        ((int4*)s_tok)[i] = ((const int4*)(tokens + row))[i];
#endif
    __syncthreads();

    // ---- Scan for earliest match of each pattern length ------------------
    const int t0 = s_pat[0], t1 = s_pat[1], t2 = s_pat[2], t3 = s_pat[3], t4 = s_pat[4];
    const int lim3 = seq - 3 - KOUT;   // p must satisfy p <= seq - n - K
    const int lim4 = seq - 4 - KOUT;
    const int lim5 = seq - 5 - KOUT;

    int m3 = IMAX, m4 = IMAX, m5 = IMAX;

    for (int base = tid * PPT; base < SEQ; base += NTHREADS * PPT) {
        const int4 lo = *(const int4*)&s_tok[base];      // ds_load_b128
        const int4 hi = *(const int4*)&s_tok[base + 4];  // ds_load_b128 (pad covers tail)
        const int w[8] = {lo.x, lo.y, lo.z, lo.w, hi.x, hi.y, hi.z, hi.w};
#pragma unroll
        for (int j = 0; j < PPT; ++j) {
            const int p = base + j;
            const bool e3 = (w[j] == t2) & (w[j + 1] == t3) & (w[j + 2] == t4);
            const bool e4 = (w[j] == t1) & (w[j + 1] == t2) & (w[j + 2] == t3) &
                            (w[j + 3] == t4);
            const bool e5 = (w[j] == t0) & (w[j + 1] == t1) & (w[j + 2] == t2) &
                            (w[j + 3] == t3) & (w[j + 4] == t4);
            if (e3 && p <= lim3 && p < m3) m3 = p;
            if (e4 && p <= lim4 && p < m4) m4 = p;
            if (e5 && p <= lim5 && p < m5) m5 = p;
        }
    }

    if (m3 != IMAX) atomicMin(&s_min[0], m3);   // ds_min on LDS
    if (m4 != IMAX) atomicMin(&s_min[1], m4);
    if (m5 != IMAX) atomicMin(&s_min[2], m5);
    __syncthreads();

    // ---- Resolve (prefer longest n) and extract K tokens -----------------
    if (tid == 0) {
        int best_n = 0, best_pos = -1;
        if      (s_min[2] != IMAX) { best_n = 5; best_pos = s_min[2]; }
        else if (s_min[1] != IMAX) { best_n = 4; best_pos = s_min[1]; }
        else if (s_min[0] != IMAX) { best_n = 3; best_pos = s_min[0]; }
        const bool has = best_pos >= 0;
        const bool msk = mask[b] != 0;
        const int  rs  = has ? best_pos + best_n : 0;
#pragma unroll
        for (int k = 0; k < KOUT; ++k) {
            int idx = rs + k;
            if (idx > SEQ - 1) idx = SEQ - 1;
            out[b * KOUT + k] = (has && msk) ? s_tok[idx] : 0;
        }
    }
}

extern "C" void kernel_launch(void* const* d_in, const int* in_sizes, int n_in,
                              void* d_out, int out_size, void* d_ws, size_t ws_size,
                              hipStream_t stream) {
    const int*           num_tokens = (const int*)d_in[0];           // (B,) int32
    const int*           tokens     = (const int*)d_in[1];           // (B,S) int32
    const unsigned char* mask       = (const unsigned char*)d_in[2]; // (B,) bool
    int*                 out        = (int*)d_out;                   // (B,K) int32
    (void)in_sizes; (void)n_in; (void)out_size; (void)d_ws; (void)ws_size;

    ngram_match_extract_kernel<<<BATCH, NTHREADS, 0, stream>>>(num_tokens, tokens, mask, out);
}